// dilate_tensor_4733053960372
// MI455X (gfx1250) — compile-verified
//
#include <hip/hip_runtime.h>
#include <hip/hip_bf16.h>

// Dilation by F=2 (row/col shift 0): out[n, 2h+1, 2w+1, c] = in[n, h, w, c],
// all other output elements are zero.
//
// Shapes (fixed by the reference):
//   in : (16, 128, 128, 128) fp32  -> 8,388,608 quads (float4)
//   out: (16, 256, 256, 128) fp32  -> 33,554,432 quads
//
// Pure streaming kernel: 128 MiB read + 512 MiB written, ~29 us floor at
// 23.3 TB/s. One thread per input quad; wave32 lanes map to the 32
// channel-quads of a pixel, so every wave-level access is a contiguous
// 512 B burst. Zero fill (3/4 of store bytes) goes through the CDNA5 async
// path: global_store_async_from_lds_b128 sourcing a 512 B zeroed LDS region
// (tracked on ASYNCcnt); the data store is a normal NT B128 store.

typedef float v4f __attribute__((ext_vector_type(4)));

__global__ __launch_bounds__(256) void dilate_f2_kernel(const v4f* __restrict__ in,
                                                        v4f* __restrict__ out) {
    __shared__ v4f zbuf[32];  // 512 B of zeros, one quad per lane
    if (threadIdx.x < 32) {
        v4f z = {0.f, 0.f, 0.f, 0.f};
        zbuf[threadIdx.x] = z;
    }
    __syncthreads();

    // Flat thread id == flat input quad index:
    //   t = ((n*128 + h)*128 + w)*32 + c4
    unsigned t  = blockIdx.x * 256u + threadIdx.x;
    unsigned c4 = t & 31u;
    unsigned w  = (t >> 5) & 127u;
    unsigned h  = (t >> 12) & 127u;
    unsigned n  = t >> 19;

    // Streaming read of the input quad (read-once -> non-temporal).
    v4f v = __builtin_nontemporal_load(in + t);

    // Output base at pixel (2h, 2w), channel-quad c4 (quad units):
    //   n stride = 256*256*32 = 2097152, h2 stride = 256*32 = 8192,
    //   w2 stride = 32.
    size_t ob = (size_t)n * 2097152u + h * 16384u + w * 64u + c4;
    v4f* p = out + ob;

    // 32-bit LDS byte address of this lane's zero quad.
    unsigned lds = (unsigned)(uintptr_t)(&zbuf[threadIdx.x & 31u]);

    // Three zero quads per thread via async store-from-LDS (ASYNCcnt path):
    //   (2h,   2w  ) -> +0 B
    //   (2h,   2w+1) -> +512 B      (32 quads)
    //   (2h+1, 2w  ) -> +131072 B   (8192 quads)
    asm volatile("global_store_async_from_lds_b128 %0, %1, off th:TH_STORE_NT"
                 :: "v"(p), "v"(lds) : "memory");
    asm volatile("global_store_async_from_lds_b128 %0, %1, off offset:512 th:TH_STORE_NT"
                 :: "v"(p), "v"(lds) : "memory");
    asm volatile("global_store_async_from_lds_b128 %0, %1, off offset:131072 th:TH_STORE_NT"
                 :: "v"(p), "v"(lds) : "memory");

    // Data quad at (2h+1, 2w+1) -> +131584 B (8224 quads), non-temporal.
    __builtin_nontemporal_store(v, p + 8224);

    // S_ENDPGM's implicit wait-idle drains ASYNCcnt before the wave retires.
}

extern "C" void kernel_launch(void* const* d_in, const int* in_sizes, int n_in,
                              void* d_out, int out_size, void* d_ws, size_t ws_size,
                              hipStream_t stream) {
    const v4f* in  = (const v4f*)d_in[0];
    v4f*       out = (v4f*)d_out;

    // 16*128*128*128 fp32 elements -> /4 = 8,388,608 quads = one per thread.
    unsigned nthreads = (unsigned)(in_sizes[0] / 4);
    dim3 block(256);
    dim3 grid(nthreads / 256u);  // 32768 blocks, exact cover, no tail
    dilate_f2_kernel<<<grid, block, 0, stream>>>(in, out);
}